// ECL_27324581937628
// MI455X (gfx1250) — compile-verified
//
#include <hip/hip_runtime.h>

// Forward of the straight-through top-k projection reduces exactly to the
// per-row hard top-k0 mask of x (see round-0 analysis). One workgroup per row:
// async-stage row to LDS, 4x8-bit radix select for the k0-th largest key,
// exact lowest-index-first tie handling, async-store the mask row.

#define TPB 256
#define MAX_DIM 4096

#if defined(__gfx1250__) &&                                          \
    __has_builtin(__builtin_amdgcn_global_load_async_to_lds_b128) && \
    __has_builtin(__builtin_amdgcn_global_store_async_from_lds_b128)
#define ECL_ASYNC_LDS 1
#else
#define ECL_ASYNC_LDS 0
#endif

// The async-LDS builtins take int32x4 pointers in AS(1)/AS(3):
//   void __builtin_amdgcn_global_load_async_to_lds_b128(v4i AS1*, v4i AS3*, Ii, Ii)
typedef int ecl_v4i __attribute__((__vector_size__(16)));
typedef __attribute__((address_space(1))) ecl_v4i* ecl_gv4p;
typedef __attribute__((address_space(3))) ecl_v4i* ecl_lv4p;

__device__ __forceinline__ ecl_gv4p ecl_to_global(const void* p) {
  return (ecl_gv4p)(__UINTPTR_TYPE__)p;
}
__device__ __forceinline__ ecl_lv4p ecl_to_lds(void* p) {
  // generic LDS pointer's low 32 bits are the wave-relative LDS address
  return (ecl_lv4p)(__UINTPTR_TYPE__)(unsigned int)(__UINTPTR_TYPE__)p;
}

__device__ __forceinline__ void ecl_wait_async0() {
#if defined(__gfx1250__)
#if __has_builtin(__builtin_amdgcn_s_wait_asynccnt)
  __builtin_amdgcn_s_wait_asynccnt(0);
#else
  asm volatile("s_wait_asynccnt 0x0" ::: "memory");
#endif
#endif
}

// order-preserving float -> u32 key (ascending key == ascending float)
__device__ __forceinline__ unsigned ecl_key(unsigned b) {
  return b ^ ((unsigned)((int)b >> 31) | 0x80000000u);
}

__global__ __launch_bounds__(TPB) void ECL_topk_mask_kernel(
    const float* __restrict__ x, const int* __restrict__ k0p,
    float* __restrict__ out, int dim) {
  __shared__ __align__(16) float rowf[MAX_DIM];  // row stage, reused for output
  __shared__ unsigned hist[256];
  __shared__ unsigned sc[257];
  __shared__ unsigned s_prefix, s_kk;

  const int t = (int)threadIdx.x;
  const int row = (int)blockIdx.x;
  const float* g = x + (size_t)row * (size_t)dim;
  float* o = out + (size_t)row * (size_t)dim;

  int k0 = k0p[0];
  if (k0 < 1) k0 = 1;
  if (k0 > dim) k0 = dim;

  // ---------- stage row into LDS (CDNA5 async global->LDS path) ----------
  const int nbytes = dim * 4;
  if ((dim & 3) == 0) {
#if ECL_ASYNC_LDS
    for (int ofs = t * 16; ofs < nbytes; ofs += TPB * 16) {
      __builtin_amdgcn_global_load_async_to_lds_b128(
          ecl_to_global((const char*)g + ofs),
          ecl_to_lds((char*)rowf + ofs), 0, 0);
    }
    ecl_wait_async0();
#else
    for (int i = t; i * 4 < dim; i += TPB)
      ((float4*)rowf)[i] = ((const float4*)g)[i];
#endif
  } else {
    for (int j = t; j < dim; j += TPB) rowf[j] = g[j];
  }
  if (t == 0) sc[256] = 0u;
  __syncthreads();

  // in-place convert to monotone keys
  unsigned* rowu = (unsigned*)rowf;
  for (int j = t; j < dim; j += TPB) rowu[j] = ecl_key(rowu[j]);
  __syncthreads();

  // ---------- 4-pass radix select: key of the k0-th largest ----------
  unsigned prefix = 0u;
  unsigned kk = (unsigned)k0;  // rank (from top) remaining among candidates
  const unsigned maskHi[4] = {0u, 0xFF000000u, 0xFFFF0000u, 0xFFFFFF00u};
  const int shifts[4] = {24, 16, 8, 0};

  for (int p = 0; p < 4; ++p) {
    hist[t] = 0u;
    __syncthreads();
    const int sh = shifts[p];
    const unsigned mh = maskHi[p];
    for (int j = t; j < dim; j += TPB) {  // strided: conflict-free LDS reads
      unsigned u = rowu[j];
      if ((u & mh) == prefix) atomicAdd(&hist[(u >> sh) & 255u], 1u);
    }
    __syncthreads();
    sc[t] = hist[t];
    __syncthreads();
#pragma unroll
    for (int off = 1; off < 256; off <<= 1) {  // inclusive suffix scan
      unsigned v = (t + off < 256) ? sc[t + off] : 0u;
      __syncthreads();
      sc[t] += v;
      __syncthreads();
    }
    unsigned ge = sc[t];      // candidates with digit >= t
    unsigned gt = sc[t + 1];  // candidates with digit >  t  (sc[256]==0)
    if (gt < kk && ge >= kk) {  // exactly one bin matches
      s_prefix = prefix | ((unsigned)t << sh);
      s_kk = kk - gt;
    }
    __syncthreads();
    prefix = s_prefix;
    kk = s_kk;
    __syncthreads();
  }
  const unsigned ut = prefix;  // threshold key (k0-th largest)
  const unsigned remT = kk;    // ties at ut to keep, lowest index first

  // ---------- tie ranks in global index order (contiguous chunks) ----------
  const int vpt = (dim + TPB - 1) / TPB;
  const int jbeg = t * vpt;
  const int jend = (jbeg + vpt < dim) ? (jbeg + vpt) : dim;

  unsigned myT = 0u;
  for (int j = jbeg; j < jend; ++j) myT += (rowu[j] == ut) ? 1u : 0u;
  sc[t] = myT;
  __syncthreads();
#pragma unroll
  for (int off = 1; off < 256; off <<= 1) {  // inclusive prefix scan
    unsigned v = (t >= off) ? sc[t - off] : 0u;
    __syncthreads();
    sc[t] += v;
    __syncthreads();
  }
  const unsigned tieBase = sc[t] - myT;  // exclusive prefix of tie counts

  // ---------- emit hard mask into LDS ----------
  unsigned seen = 0u;
  for (int j = jbeg; j < jend; ++j) {
    unsigned u = rowu[j];
    float y;
    if (u > ut) {
      y = 1.0f;
    } else if (u == ut) {
      y = (tieBase + seen < remT) ? 1.0f : 0.0f;
      ++seen;
    } else {
      y = 0.0f;
    }
    rowf[j] = y;
  }
  __syncthreads();

  // ---------- store row (CDNA5 async LDS->global path) ----------
  if ((dim & 3) == 0) {
#if ECL_ASYNC_LDS
    for (int ofs = t * 16; ofs < nbytes; ofs += TPB * 16) {
      __builtin_amdgcn_global_store_async_from_lds_b128(
          ecl_to_global((char*)o + ofs),
          ecl_to_lds((char*)rowf + ofs), 0, 0);
    }
    ecl_wait_async0();
#else
    for (int i = t; i * 4 < dim; i += TPB)
      ((float4*)o)[i] = ((const float4*)rowf)[i];
#endif
  } else {
    for (int j = t; j < dim; j += TPB) o[j] = rowf[j];
  }
}

extern "C" void kernel_launch(void* const* d_in, const int* in_sizes, int n_in,
                              void* d_out, int out_size, void* d_ws,
                              size_t ws_size, hipStream_t stream) {
  (void)n_in;
  (void)d_ws;
  (void)ws_size;
  (void)out_size;
  const float* x = (const float*)d_in[0];
  const int* k0 = (const int*)d_in[2];  // d_in[1] (k per row) unused in forward
  float* out = (float*)d_out;

  const int bs = in_sizes[1];                 // k has shape [bs]
  const int dim = (bs > 0) ? in_sizes[0] / bs : 0;
  if (bs <= 0 || dim <= 0 || dim > MAX_DIM) return;

  ECL_topk_mask_kernel<<<bs, TPB, 0, stream>>>(x, k0, out, dim);
}